// TopGNNModule_53575422050968
// MI455X (gfx1250) — compile-verified
//
#include <hip/hip_runtime.h>
#include <hip/hip_bf16.h>

// Problem constants (match reference)
#define BB 16
#define SS 512
#define HH 12
#define DD 768
#define PP 256
#define KK 51

typedef __attribute__((ext_vector_type(2))) float v2f;
typedef __attribute__((ext_vector_type(8))) float v8f;

// ---------------------------------------------------------------------------
// Kernel 1: average attention over heads for one (b,t) row, bitonic-sort the
// 512 values descending (ties -> smaller index first, matching lax.top_k),
// emit top-51 indices + count of kept (positive-prefix; keep-all fallback).
// One block of 256 threads per row; 8192 blocks. Reads 201 MB total (HBM bound).
// ---------------------------------------------------------------------------
__global__ void __launch_bounds__(256)
avg_topk_kernel(const float* __restrict__ attn,
                int* __restrict__ idx_out,
                int* __restrict__ cnt_out) {
    __shared__ float sv[SS];
    __shared__ int   si[SS];
    __shared__ int   scnt;

    const int row = blockIdx.x;            // b*S + t
    const int b   = row >> 9;
    const int t   = row & (SS - 1);
    const int tid = threadIdx.x;           // 0..255

    // head-averaged row: coalesced across columns, 12 strided head planes
    for (int c = tid; c < SS; c += 256) {
        const float* p = attn + (((size_t)b * HH) * SS + t) * SS + c;
        float s = 0.0f;
        #pragma unroll
        for (int h = 0; h < HH; ++h) s += p[(size_t)h * SS * SS];
        sv[c] = s * (1.0f / (float)HH);
        si[c] = c;
    }
    __syncthreads();

    // bitonic sort, 512 elements, 256 compare-exchanges per stage
    for (int k = 2; k <= SS; k <<= 1) {
        for (int j = k >> 1; j > 0; j >>= 1) {
            const int i = ((tid & ~(j - 1)) << 1) | (tid & (j - 1));
            const int p = i | j;
            const bool dirDesc = ((i & k) == 0);
            float va = sv[i], vb = sv[p];
            int   ia = si[i], ib = si[p];
            // "a ranks before b" in descending order with index tiebreak
            const bool aFirst = (va > vb) || (va == vb && ia < ib);
            const bool doSwap = dirDesc ? !aFirst : aFirst;
            if (doSwap) { sv[i] = vb; sv[p] = va; si[i] = ib; si[p] = ia; }
            __syncthreads();
        }
    }

    // sorted descending -> positives form a prefix of the top-K
    if (tid == 0) {
        int c = 0;
        for (int j = 0; j < KK; ++j) {
            if (sv[j] > 0.0f) ++c; else break;
        }
        scnt = (c == 0) ? KK : c;   // no positive -> uniform softmax -> keep all
    }
    __syncthreads();
    if (tid < KK) idx_out[(size_t)row * KK + tid] = si[tid];
    if (tid == 0) cnt_out[row] = scnt;
}

// ---------------------------------------------------------------------------
// Kernel 2: h = hidden_state @ W_node + b_node  via V_WMMA_F32_16X16X4_F32.
// M=8192, N=256, K=768. One 16x16 C tile per wave (8 waves / block).
// A frag (16x4 f32): lanes 0-15 M=0..15 K={k,k+1}; lanes 16-31 K={k+2,k+3}.
// B frag (4x16 f32): VGPR0 rows K={k | k+2}, VGPR1 rows K={k+1 | k+3}, N=lane%16.
// ---------------------------------------------------------------------------
__global__ void __launch_bounds__(256)
node_gemm_wmma_kernel(const float* __restrict__ A,      // [8192, 768]
                      const float* __restrict__ W,      // [768, 256]
                      const float* __restrict__ bias,   // [256]
                      float* __restrict__ C) {          // [8192, 256]
    const int lane  = threadIdx.x & 31;
    const int wave  = threadIdx.x >> 5;
    const int tile  = blockIdx.x * 8 + wave;   // 8192 tiles total
    const int tn    = tile & 15;               // 256/16 = 16 N-tiles
    const int tm    = tile >> 4;               // 512 M-tiles
    const int m0    = tm << 4;
    const int n0    = tn << 4;
    const int half  = lane >> 4;               // 0: lanes 0-15, 1: lanes 16-31
    const int l16   = lane & 15;

    v8f acc = {};
    const float* arow = A + (size_t)(m0 + l16) * DD;
    for (int k = 0; k < DD; k += 4) {
        v2f a, bf;
        const float* ap = arow + k + half * 2;
        a.x = ap[0];
        a.y = ap[1];
        const float* bp = W + (size_t)(k + half * 2) * PP + n0 + l16;
        bf.x = bp[0];
        bf.y = bp[PP];
        acc = __builtin_amdgcn_wmma_f32_16x16x4_f32(
            /*neg_a=*/false, a, /*neg_b=*/false, bf,
            /*c_mod=*/(short)0, acc, /*reuse_a=*/false, /*reuse_b=*/false);
    }

    // C layout: VGPR r holds M = r + half*8, N = l16
    const float bv = bias[n0 + l16];
    #pragma unroll
    for (int r = 0; r < 8; ++r) {
        C[(size_t)(m0 + r + half * 8) * PP + n0 + l16] = acc[r] + bv;
    }
}

// ---------------------------------------------------------------------------
// Kernel 3: per (b, group-of-16 t): h_agg = mean over kept top-k of h[b,idx,:],
// new_h = eta*h + (1-eta)*h_agg, accumulate over the 16 t's -> partial sums.
// Deterministic (no fp atomics). Gathers hit L2 (h per batch = 512 KB).
// ---------------------------------------------------------------------------
__global__ void __launch_bounds__(256)
gather_agg_kernel(const float* __restrict__ h,        // [8192, 256]
                  const int* __restrict__ idx_buf,    // [8192, 51]
                  const int* __restrict__ cnt_buf,    // [8192]
                  const float* __restrict__ eta_p,    // [1]
                  float* __restrict__ partial) {      // [B*32, 256]
    const int ch  = threadIdx.x;                      // 0..255 channel
    const int g   = blockIdx.x;                       // b*32 + group
    const int b   = g >> 5;
    const int grp = g & 31;
    const float eta  = eta_p[0];
    const float ceta = 1.0f - eta;

    float acc = 0.0f;
    for (int tt = 0; tt < 16; ++tt) {
        const int t   = (grp << 4) + tt;
        const int row = (b << 9) + t;
        const int cnt = cnt_buf[row];
        const int* ip = idx_buf + (size_t)row * KK;
        float s = 0.0f;
        for (int j = 0; j < cnt; ++j) {
            const int src = ip[j];                    // uniform across block
            s += h[((size_t)(b << 9) + src) * PP + ch];
        }
        const float hagg = s / (float)cnt;
        const float hown = h[(size_t)row * PP + ch];
        acc += eta * hown + ceta * hagg;
    }
    partial[(size_t)g * PP + ch] = acc;
}

// ---------------------------------------------------------------------------
// Kernel 4 (per batch): reduce 32 partials -> mean over S -> tanh -> fc ->
// LayerNorm. One block of 256 threads per batch element.
// ---------------------------------------------------------------------------
__global__ void __launch_bounds__(256)
finalize_kernel(const float* __restrict__ partial,    // [B*32, 256]
                const float* __restrict__ W_fc,       // [256, 256]
                const float* __restrict__ b_fc,       // [256]
                const float* __restrict__ gamma,      // [256]
                const float* __restrict__ beta,       // [256]
                float* __restrict__ out) {            // [B, 256]
    __shared__ float v[PP];
    __shared__ float red[PP];

    const int b   = blockIdx.x;
    const int tid = threadIdx.x;

    float s = 0.0f;
    for (int g = 0; g < 32; ++g)
        s += partial[((size_t)(b * 32 + g)) * PP + tid];
    v[tid] = tanhf(s * (1.0f / (float)SS));
    __syncthreads();

    // fc: out2[tid] = sum_c v[c] * W_fc[c, tid] + b_fc[tid]
    float dot = b_fc[tid];
    for (int c = 0; c < PP; ++c)
        dot += v[c] * W_fc[(size_t)c * PP + tid];

    // LayerNorm over 256 channels
    red[tid] = dot;
    __syncthreads();
    for (int off = 128; off > 0; off >>= 1) {
        if (tid < off) red[tid] += red[tid + off];
        __syncthreads();
    }
    const float mu = red[0] * (1.0f / (float)PP);
    __syncthreads();
    const float d = dot - mu;
    red[tid] = d * d;
    __syncthreads();
    for (int off = 128; off > 0; off >>= 1) {
        if (tid < off) red[tid] += red[tid + off];
        __syncthreads();
    }
    const float var = red[0] * (1.0f / (float)PP);
    out[(size_t)b * PP + tid] = d * rsqrtf(var + 1e-5f) * gamma[tid] + beta[tid];
}

// ---------------------------------------------------------------------------
// Launch
// ---------------------------------------------------------------------------
extern "C" void kernel_launch(void* const* d_in, const int* in_sizes, int n_in,
                              void* d_out, int out_size, void* d_ws, size_t ws_size,
                              hipStream_t stream) {
    const float* hidden = (const float*)d_in[0];   // [16,512,768]
    const float* attn   = (const float*)d_in[1];   // [16,12,512,512]
    // d_in[2]: length (== S, unused)
    const float* W_node = (const float*)d_in[3];   // [768,256]
    const float* b_node = (const float*)d_in[4];   // [256]
    const float* eta    = (const float*)d_in[5];   // [1]
    const float* W_fc   = (const float*)d_in[6];   // [256,256]
    const float* b_fc   = (const float*)d_in[7];   // [256]
    const float* gamma  = (const float*)d_in[8];   // [256]
    const float* beta   = (const float*)d_in[9];   // [256]
    float* out = (float*)d_out;

    // workspace layout (256B-aligned slabs)
    char* ws = (char*)d_ws;
    const size_t idx_bytes = (size_t)BB * SS * KK * sizeof(int);      // 1,671,168
    const size_t cnt_bytes = (size_t)BB * SS * sizeof(int);           //    32,768
    const size_t h_bytes   = (size_t)BB * SS * PP * sizeof(float);    // 8,388,608
    size_t o = 0;
    int*   idx_buf = (int*)(ws + o);   o += (idx_bytes + 255) & ~(size_t)255;
    int*   cnt_buf = (int*)(ws + o);   o += (cnt_bytes + 255) & ~(size_t)255;
    float* h_buf   = (float*)(ws + o); o += (h_bytes + 255) & ~(size_t)255;
    float* partial = (float*)(ws + o); // [B*32, 256] = 524,288 B
    (void)ws_size; (void)in_sizes; (void)n_in; (void)out_size;

    // 1) head-average + top-k per row: 8192 rows
    avg_topk_kernel<<<BB * SS, 256, 0, stream>>>(attn, idx_buf, cnt_buf);

    // 2) node_trans GEMM via fp32 WMMA: 8192 tiles / 8 waves per block
    node_gemm_wmma_kernel<<<(BB * SS / 16) * (PP / 16) / 8, 256, 0, stream>>>(
        hidden, W_node, b_node, h_buf);

    // 3) gather + per-group deterministic partial reduction: B*32 blocks
    gather_agg_kernel<<<BB * 32, 256, 0, stream>>>(h_buf, idx_buf, cnt_buf, eta, partial);

    // 4) per-batch finalize: mean -> tanh -> fc -> LayerNorm
    finalize_kernel<<<BB, 256, 0, stream>>>(partial, W_fc, b_fc, gamma, beta, out);
}